// GraphRandomNetwork_3083786519266
// MI455X (gfx1250) — compile-verified
//
#include <hip/hip_runtime.h>
#include <hip/hip_bf16.h>

// ---------------------------------------------------------------------------
// GraphRandomNetwork: batched synchronous local-push PPR (alpha=0.5) +
// top-k sparsify + feature propagation, for MI455X (gfx1250).
//
//  * fp32 WMMA (V_WMMA_F32_16X16X4_F32): the 1e-5 residual threshold forbids
//    low-precision accumulation, and fp32 is the reference dtype.
//  * 2x2 register blocking per wave (32x32 tile, 4 independent accumulators)
//    -> halves LDS fragment traffic per WMMA and hides WMMA->WMMA hazards.
//  * B-operand tiles staged with GLOBAL_LOAD_ASYNC_TO_LDS (ASYNCcnt) —
//    builtin prototype: (AS1 int*, AS3 int*, imm offset, imm cpol).
//    A-operand staged through VGPRs so the push mask fuses in.
//  * Fixed 48 iterations + device-side convergence flag chain (capture-safe;
//    converged iterations are exact no-ops that early-exit).
//  * A, R, P, T (4 x 16 MB) live in d_ws -> resident in the 192 MB L2.
// ---------------------------------------------------------------------------

#define NNODES   2048
#define NEDGES   32768
#define RMAXC    1e-05f
#define TOPK     20
#define MAX_ITERS 48

#define BM 128           // block tile rows
#define BN 64            // block tile cols
#define BK 32            // K slab staged in LDS
#define BPAD 8           // ldsB row pad: 2*(BN+BPAD) % 64 == 16 -> the two
                         // lane-halves of a B-frag hit disjoint bank groups

typedef float v2f __attribute__((ext_vector_type(2)));
typedef float v8f __attribute__((ext_vector_type(8)));

typedef __attribute__((address_space(1))) int* as1_i32p;
typedef __attribute__((address_space(3))) int* as3_i32p;

// Async LDS staging path (gfx1250): use only if the toolchain declares it.
#if defined(__has_builtin)
#if __has_builtin(__builtin_amdgcn_global_load_async_to_lds_b32) && \
    __has_builtin(__builtin_amdgcn_s_wait_asynccnt)
#define USE_ASYNC_LDS 1
#endif
#endif
#ifndef USE_ASYNC_LDS
#define USE_ASYNC_LDS 0
#endif

// ---------------------------------------------------------------------------
// Init: A = 0, P = 0, R = I, flags[0]=1 (R=I has entries >= RMAX), rest 0.
// ---------------------------------------------------------------------------
__global__ void grn_init_kernel(float* __restrict__ A, float* __restrict__ R,
                                float* __restrict__ P, int* __restrict__ flags) {
    int i = blockIdx.x * blockDim.x + threadIdx.x;
    if (i < NNODES * NNODES) {
        A[i] = 0.0f;
        P[i] = 0.0f;
        R[i] = ((i >> 11) == (i & (NNODES - 1))) ? 1.0f : 0.0f;
    }
    if (i <= MAX_ITERS) flags[i] = (i == 0) ? 1 : 0;
}

// ---------------------------------------------------------------------------
// Scatter edges: A[row, col] += 1 (duplicates summed, matches csr_matrix).
// ---------------------------------------------------------------------------
__global__ void grn_scatter_kernel(const int* __restrict__ ei, float* __restrict__ A) {
    int e = blockIdx.x * blockDim.x + threadIdx.x;
    if (e < NEDGES) {
        int r = ei[e];
        int c = ei[NEDGES + e];
        atomicAdd(&A[(size_t)r * NNODES + c], 1.0f);
    }
}

// ---------------------------------------------------------------------------
// Row-normalize: A[row,:] /= max(deg,1) with true division (matches jnp).
// ---------------------------------------------------------------------------
__global__ void grn_rownorm_kernel(float* __restrict__ A) {
    __shared__ float sred[256];
    int row = blockIdx.x;
    float* a = A + (size_t)row * NNODES;
    float s = 0.0f;
    for (int c = threadIdx.x; c < NNODES; c += 256) s += a[c];
    sred[threadIdx.x] = s;
    __syncthreads();
    for (int off = 128; off > 0; off >>= 1) {
        if (threadIdx.x < off) sred[threadIdx.x] += sred[threadIdx.x + off];
        __syncthreads();
    }
    float deg = sred[0];
    deg = (deg == 0.0f) ? 1.0f : deg;
    for (int c = threadIdx.x; c < NNODES; c += 256) a[c] = a[c] / deg;
}

// ---------------------------------------------------------------------------
// fp32 WMMA GEMM:  C = maskedA(Asrc) @ Bsrc   (2048^3)
//   applyMask != 0:  a -> (a >= RMAX) ? a : 0   (push mask, fused in staging)
//   runflag: if non-null and *runflag == 0 -> converged, no-op.
//
// Block = 256 threads = 8 wave32s in a 4x2 grid; each wave owns a 32x32
// output tile = 2x2 WMMA subtiles (4 independent accumulator chains).
// K staged in 32-wide LDS slabs; 32 V_WMMA_F32_16X16X4_F32 per slab.
//
// 16x16x4 f32 fragment layout (ISA 7.12.2):
//   A (16x4): lane L<16 -> M=L, {K=0,1}; lane L>=16 -> M=L-16, {K=2,3}.
//   B (4x16): mirrored; N = lane%16.
//   C/D: VGPR v holds M=v (lanes 0-15) / M=v+8 (lanes 16-31), N = lane%16.
// ---------------------------------------------------------------------------
__launch_bounds__(256)
__global__ void grn_wmma_gemm_kernel(const float* __restrict__ Asrc,
                                     const float* __restrict__ Bsrc,
                                     float* __restrict__ C,
                                     const int* runflag, int applyMask) {
    if (runflag && *runflag == 0) return;   // uniform branch: EXEC stays full

    __shared__ float ldsA[BM][BK + 1];      // 128 x 33 (odd stride)
    __shared__ float ldsB[BK][BN + BPAD];   // 32 x 72

    const int tid  = threadIdx.x;
    const int wave = tid >> 5;
    const int lane = tid & 31;
    const int lr   = lane & 15;             // 0..15
    const int lh   = lane >> 4;             // lane half: 0 or 1

    const int row0 = blockIdx.y * BM;
    const int col0 = blockIdx.x * BN;
    const int m0   = (wave & 3) * 32;       // wave tile row inside block
    const int n0   = (wave >> 2) * 32;      // wave tile col inside block

    v8f acc00 = {0.f,0.f,0.f,0.f,0.f,0.f,0.f,0.f};
    v8f acc01 = acc00, acc10 = acc00, acc11 = acc00;

    for (int k0 = 0; k0 < NNODES; k0 += BK) {
        // --- Stage A tile (128 x 32) through VGPRs, push mask fused -------
#pragma unroll
        for (int i = 0; i < 16; ++i) {
            int idx = tid + i * 256;
            int r = idx >> 5, c = idx & 31;
            float v = Asrc[(size_t)(row0 + r) * NNODES + (k0 + c)];
            if (applyMask) v = (v >= RMAXC) ? v : 0.0f;
            ldsA[r][c] = v;
        }
        // --- Stage B tile (32 x 64): async direct-to-LDS (ASYNCcnt) -------
#if USE_ASYNC_LDS
#pragma unroll
        for (int i = 0; i < 8; ++i) {
            int idx = tid + i * 256;
            int r = idx >> 6, c = idx & 63;
            __builtin_amdgcn_global_load_async_to_lds_b32(
                (as1_i32p)(Bsrc + (size_t)(k0 + r) * NNODES + (col0 + c)),
                (as3_i32p)&ldsB[r][c],
                0, 0);
        }
#else
#pragma unroll
        for (int i = 0; i < 8; ++i) {
            int idx = tid + i * 256;
            int r = idx >> 6, c = idx & 63;
            ldsB[r][c] = Bsrc[(size_t)(k0 + r) * NNODES + (col0 + c)];
        }
#endif
        // --- Prefetch next K slab (global_prefetch_b8) --------------------
        if (k0 + BK < NNODES) {
            if (tid < BM) {         // one 128B line covers a 32-float A row
                __builtin_prefetch(&Asrc[(size_t)(row0 + tid) * NNODES + (k0 + BK)], 0, 1);
            } else if (tid < BM + 2 * BK) {   // 2 lines per 64-float B row
                int t = tid - BM;
                __builtin_prefetch(&Bsrc[(size_t)(k0 + BK + (t >> 1)) * NNODES
                                         + col0 + (t & 1) * 32], 0, 1);
            }
        }
#if USE_ASYNC_LDS
        __builtin_amdgcn_s_wait_asynccnt(0);
#endif
        __syncthreads();

        // --- 2x2 register-blocked WMMA over the slab ----------------------
#pragma unroll
        for (int kk = 0; kk < BK; kk += 4) {
            v2f a0, a1, b0, b1;
            a0.x = ldsA[m0 + lr][kk + 2 * lh];
            a0.y = ldsA[m0 + lr][kk + 2 * lh + 1];
            a1.x = ldsA[m0 + 16 + lr][kk + 2 * lh];
            a1.y = ldsA[m0 + 16 + lr][kk + 2 * lh + 1];
            b0.x = ldsB[kk + 2 * lh][n0 + lr];
            b0.y = ldsB[kk + 2 * lh + 1][n0 + lr];
            b1.x = ldsB[kk + 2 * lh][n0 + 16 + lr];
            b1.y = ldsB[kk + 2 * lh + 1][n0 + 16 + lr];
            // 8 args: (neg_a, A, neg_b, B, c_mod, C, reuse_a, reuse_b)
            acc00 = __builtin_amdgcn_wmma_f32_16x16x4_f32(
                false, a0, false, b0, (short)0, acc00, false, false);
            acc01 = __builtin_amdgcn_wmma_f32_16x16x4_f32(
                false, a0, false, b1, (short)0, acc01, false, false);
            acc10 = __builtin_amdgcn_wmma_f32_16x16x4_f32(
                false, a1, false, b0, (short)0, acc10, false, false);
            acc11 = __builtin_amdgcn_wmma_f32_16x16x4_f32(
                false, a1, false, b1, (short)0, acc11, false, false);
        }
        __syncthreads();
    }

#pragma unroll
    for (int v = 0; v < 8; ++v) {
        int rA = row0 + m0 + v + 8 * lh;        // subtile rows 0..15
        int rB = rA + 16;                       // subtile rows 16..31
        int cA = col0 + n0 + lr;
        C[(size_t)rA * NNODES + cA]      = acc00[v];
        C[(size_t)rA * NNODES + cA + 16] = acc01[v];
        C[(size_t)rB * NNODES + cA]      = acc10[v];
        C[(size_t)rB * NNODES + cA + 16] = acc11[v];
    }
}

// ---------------------------------------------------------------------------
// Push update:  push = R * 1[R>=RMAX];  P += 0.5*push;
//               R = R - push + 0.5*T.  flags[it+1] = any(R >= RMAX).
// ---------------------------------------------------------------------------
__global__ void grn_update_kernel(float* __restrict__ R, const float* __restrict__ T,
                                  float* __restrict__ P,
                                  const int* __restrict__ flagIn,
                                  int* __restrict__ flagOut) {
    if (*flagIn == 0) return;               // converged: iteration is a no-op
    __shared__ int sAny;
    if (threadIdx.x == 0) sAny = 0;
    __syncthreads();

    size_t i = ((size_t)blockIdx.x * blockDim.x + threadIdx.x) * 4;
    float4 r = *(const float4*)(R + i);
    float4 t = *(const float4*)(T + i);
    float4 p = *(const float4*)(P + i);
    float* rr = (float*)&r;
    float* tt = (float*)&t;
    float* pp = (float*)&p;
    int any = 0;
#pragma unroll
    for (int c = 0; c < 4; ++c) {
        float push = (rr[c] >= RMAXC) ? rr[c] : 0.0f;
        pp[c] += 0.5f * push;
        rr[c] = rr[c] - push + 0.5f * tt[c];
        any |= (rr[c] >= RMAXC) ? 1 : 0;
    }
    *(float4*)(R + i) = r;
    *(float4*)(P + i) = p;

    if (any) atomicOr(&sAny, 1);
    __syncthreads();
    if (threadIdx.x == 0 && sAny) atomicOr(flagOut, 1);
}

// ---------------------------------------------------------------------------
// Top-k sparsify (k=20) per row; ties -> lowest column index (lax.top_k).
// One block per row; iterative block-wide argmax over an LDS row buffer.
// ---------------------------------------------------------------------------
__global__ void grn_topk_kernel(const float* __restrict__ P, float* __restrict__ Pout) {
    __shared__ float buf[NNODES];
    __shared__ float rv[256];
    __shared__ int   ri[256];
    __shared__ float selv[TOPK];
    __shared__ int   seli[TOPK];

    int row = blockIdx.x;
    const float* src = P + (size_t)row * NNODES;
    for (int c = threadIdx.x; c < NNODES; c += 256) buf[c] = src[c];
    __syncthreads();

    for (int t = 0; t < TOPK; ++t) {
        float bv = -1.0f;                    // P entries are >= 0
        int   bi = NNODES;
        for (int c = threadIdx.x; c < NNODES; c += 256) {
            float v = buf[c];
            if (v > bv) { bv = v; bi = c; }  // ascending scan -> lowest index on ties
        }
        rv[threadIdx.x] = bv;
        ri[threadIdx.x] = bi;
        __syncthreads();
        for (int off = 128; off > 0; off >>= 1) {
            if (threadIdx.x < off) {
                float ov = rv[threadIdx.x + off];
                int   oi = ri[threadIdx.x + off];
                if (ov > rv[threadIdx.x] ||
                    (ov == rv[threadIdx.x] && oi < ri[threadIdx.x])) {
                    rv[threadIdx.x] = ov;
                    ri[threadIdx.x] = oi;
                }
            }
            __syncthreads();
        }
        if (threadIdx.x == 0) {
            selv[t] = rv[0];
            seli[t] = ri[0];
            buf[ri[0]] = -2.0f;              // remove from future rounds
        }
        __syncthreads();
    }

    float* dst = Pout + (size_t)row * NNODES;
    for (int c = threadIdx.x; c < NNODES; c += 256) dst[c] = 0.0f;
    __syncthreads();
    if (threadIdx.x < TOPK) dst[seli[threadIdx.x]] = selv[threadIdx.x];
}

// ---------------------------------------------------------------------------
// kernel_launch
//   d_in[0]: node_feats  (2048*2048 f32)
//   d_in[1]: edge_index  (2*32768 i32)
//   d_out  : latent      (2048*2048 f32)
//   d_ws   : A | R | P | T(+Ptopk) = 4 * 16 MB  + flag array
// ---------------------------------------------------------------------------
extern "C" void kernel_launch(void* const* d_in, const int* in_sizes, int n_in,
                              void* d_out, int out_size, void* d_ws, size_t ws_size,
                              hipStream_t stream) {
    (void)in_sizes; (void)n_in; (void)out_size; (void)ws_size;

    const float* node_feats = (const float*)d_in[0];
    const int*   edge_index = (const int*)d_in[1];
    float*       out        = (float*)d_out;

    const size_t NN = (size_t)NNODES * NNODES;
    float* A     = (float*)d_ws;           // normalized adjacency
    float* R     = A + NN;                 // residual
    float* P     = R + NN;                 // PPR accumulator
    float* T     = P + NN;                 // push @ A scratch; reused as P_topk
    int*   flags = (int*)(T + NN);         // MAX_ITERS+1 convergence flags

    dim3 blk(256);
    dim3 gridElem((unsigned)((NN + 255) / 256));
    dim3 gridGemm(NNODES / BN, NNODES / BM);   // (32, 16) blocks

    // 1) init A=0, P=0, R=I, flags
    grn_init_kernel<<<gridElem, blk, 0, stream>>>(A, R, P, flags);

    // 2) build + normalize adjacency
    grn_scatter_kernel<<<dim3(NEDGES / 256), blk, 0, stream>>>(edge_index, A);
    grn_rownorm_kernel<<<dim3(NNODES), blk, 0, stream>>>(A);

    // 3) push loop: fixed trip count with device-side convergence short-circuit
    for (int it = 0; it < MAX_ITERS; ++it) {
        // T = push(R) @ A   (mask fused into the A-operand staging)
        grn_wmma_gemm_kernel<<<gridGemm, blk, 0, stream>>>(R, A, T, &flags[it], 1);
        // P += 0.5*push; R = R - push + 0.5*T; flags[it+1] = any(R >= RMAX)
        grn_update_kernel<<<dim3((unsigned)(NN / 4 / 256)), blk, 0, stream>>>(
            R, T, P, &flags[it], &flags[it + 1]);
    }

    // 4) top-20 per row into T (reused as sparsified P)
    grn_topk_kernel<<<dim3(NNODES), blk, 0, stream>>>(P, T);

    // 5) latent = P_topk @ node_feats  (dense fp32 WMMA GEMM, no mask)
    grn_wmma_gemm_kernel<<<gridGemm, blk, 0, stream>>>(T, node_feats, out,
                                                       (const int*)nullptr, 0);
}